// MountainCar1_80187039416384
// MI455X (gfx1250) — compile-verified
//
#include <hip/hip_runtime.h>

typedef __attribute__((ext_vector_type(2))) float v2f;
typedef __attribute__((ext_vector_type(8))) float v8f;

#define MC_MIN_P   (-1.2f)
#define MC_GOAL    (0.5f)
#define MC_UTHRESH (0.5f)

__device__ __forceinline__ float xor16_add(float x) {
#if __has_builtin(__builtin_amdgcn_permlanex16)
    // lane i <-> lane i+16 exchange, pure VALU (identity lane selects).
    unsigned o = __builtin_amdgcn_permlanex16(
        __float_as_uint(x), __float_as_uint(x),
        0x76543210u, 0xfedcba98u, false, false);
    return x + __uint_as_float(o);
#else
    return x + __shfl_xor(x, 16, 32);
#endif
}

// Hardware tanh (CDNA5 TRANS op). TRANS results need 1 independent op before
// use and HW does not interlock co-execution for raw asm, so bundle a v_nop.
__device__ __forceinline__ float fast_tanh(float x) {
    float r;
    asm("v_tanh_f32 %0, %1\n\tv_nop" : "=v"(r) : "v"(x));
    return r;
}

// Single-instruction relu: raw v_max_num_f32 avoids the canonicalize op that
// fmaxf()/select lowering inserts before IEEE maxnum.
__device__ __forceinline__ float relu_f(float x) {
    float r;
    asm("v_max_num_f32 %0, 0, %1" : "=v"(r) : "v"(x));
    return r;
}

// One wave32 handles 16 agents. WMMA computes H^T = W1^T (64x2, K padded to 4)
// @ X^T (4x16 agents) as 4 tiles of 16 hidden rows. b1 rides in the C operand.
__global__ __launch_bounds__(256) void mountaincar_wmma_kernel(
    const float* __restrict__ x,      // (B,4) [p,v,u,a]
    const float* __restrict__ W1,     // (2,64)
    const float* __restrict__ b1,     // (64)
    const float* __restrict__ W2,     // (64,1)
    const float* __restrict__ b2,     // (1)
    const int*   __restrict__ n_steps_p,
    float* __restrict__ out,          // (B,4)
    int B)
{
    const int lane        = threadIdx.x & 31;
    const int waveInBlock = threadIdx.x >> 5;
    const int wave        = blockIdx.x * (blockDim.x >> 5) + waveInBlock;
    const int laneA       = lane & 15;          // agent sub-index within wave
    const bool hiHalf     = lane >= 16;
    const int agent       = wave * 16 + laneA;
    if (agent >= B) return;                     // B multiple of 128; never taken

    const int T = *n_steps_p;

    // ---- A tiles: W1^T, 4 tiles of 16 hidden rows. K=2,3 (hi half) are kept
    // exactly 0.0 so the hi-half B rows never need per-step zeroing
    // (0 * finite == 0; state values are always finite).
    v2f a[4];
#pragma unroll
    for (int t = 0; t < 4; ++t) {
        const int m = t * 16 + laneA;
        a[t].x = (!hiHalf) ? W1[0 * 64 + m] : 0.0f;   // K=0 -> weight on p
        a[t].y = (!hiHalf) ? W1[1 * 64 + m] : 0.0f;   // K=1 -> weight on v
    }

    // ---- Per-lane view of the 8 D-rows this lane produces per tile:
    // lanes 0-15 see rows 0-7, lanes 16-31 see rows 8-15 (C/D layout).
    const int rowOff = hiHalf ? 8 : 0;
    float w2l[4][8];
    v8f   cbias[4];
#pragma unroll
    for (int t = 0; t < 4; ++t) {
#pragma unroll
        for (int r = 0; r < 8; ++r) {
            const int h = t * 16 + rowOff + r;
            w2l[t][r]   = W2[h];
            cbias[t][r] = b1[h];                // C[M][N] = b1[M] for all N
        }
    }
    const float bias2 = b2[0];

    // ---- Initial state (lane halves duplicate the same agent; stays in sync).
    float p  = x[agent * 4 + 0];
    float v  = x[agent * 4 + 1];
    float u  = x[agent * 4 + 2];
    float av = x[agent * 4 + 3];

    for (int step = 0; step < T; ++step) {
        const bool active = (p <= MC_GOAL);
        const bool reset  = (p <= MC_MIN_P);
        const float pr = reset ? MC_MIN_P : p;
        const float vr = reset ? 0.0f : v;

        // B-matrix 4x16: lanes 0-15 carry K=0,1 rows = [pr, vr]. Hi-half lanes
        // carry K=2,3 rows; those multiply against A's exact zeros, so no
        // per-step masking is needed.
        v2f bmat;
        bmat.x = pr;
        bmat.y = vr;

        // ---- Issue all 4 WMMAs with independent dests so their latency
        // overlaps the VALU reduction.
        v8f d[4];
#pragma unroll
        for (int t = 0; t < 4; ++t) {
            d[t] = __builtin_amdgcn_wmma_f32_16x16x4_f32(
                /*neg_a=*/false, a[t], /*neg_b=*/false, bmat,
                /*c_mod=*/(short)0, cbias[t],
                /*reuse_a=*/false, /*reuse_b=*/false);
        }

        // ---- Layer 2: 8 independent accumulator chains (one per row index),
        // each of depth 4 (tiles); compiler packs pairs into v_pk_fma_f32.
        float acc[8];
#pragma unroll
        for (int r = 0; r < 8; ++r)
            acc[r] = relu_f(d[0][r]) * w2l[0][r];
#pragma unroll
        for (int t = 1; t < 4; ++t)
#pragma unroll
            for (int r = 0; r < 8; ++r)
                acc[r] = fmaf(relu_f(d[t][r]), w2l[t][r], acc[r]);

        float partial = ((acc[0] + acc[1]) + (acc[2] + acc[3]))
                      + ((acc[4] + acc[5]) + (acc[6] + acc[7]));

        // Combine the two lane-halves (rows 0-7 + rows 8-15 of each tile).
        const float total = xor16_add(partial);

        const float an = fast_tanh(total + bias2);
        const float un = (an <= MC_UTHRESH) ? -1.0f : 1.0f;
        const float vn = vr + un * 0.0015f - 0.0025f * __cosf(3.0f * pr);
        const float pn = pr + vn;

        p  = active ? pn : p;
        v  = active ? vn : v;
        u  = active ? un : u;
        av = active ? an : av;
    }

    if (!hiHalf) {
        out[agent * 4 + 0] = p;
        out[agent * 4 + 1] = v;
        out[agent * 4 + 2] = u;
        out[agent * 4 + 3] = av;
    }
}

extern "C" void kernel_launch(void* const* d_in, const int* in_sizes, int n_in,
                              void* d_out, int out_size, void* d_ws, size_t ws_size,
                              hipStream_t stream) {
    (void)n_in; (void)d_ws; (void)ws_size; (void)out_size;
    const float* x   = (const float*)d_in[0];   // (B,4)
    const float* W1  = (const float*)d_in[1];   // (2,64)
    const float* b1  = (const float*)d_in[2];   // (64)
    const float* W2  = (const float*)d_in[3];   // (64,1)
    const float* b2  = (const float*)d_in[4];   // (1)
    const int* nstp  = (const int*)d_in[5];     // scalar
    float* out       = (float*)d_out;

    const int B = in_sizes[0] / 4;              // 65536
    // 16 agents per wave, 8 waves per 256-thread block -> 128 agents per block.
    const int blocks = (B + 127) / 128;
    mountaincar_wmma_kernel<<<blocks, 256, 0, stream>>>(x, W1, b1, W2, b2, nstp, out, B);
}